// BlockReLU_V1_1357209666244
// MI455X (gfx1250) — compile-verified
//
#include <hip/hip_runtime.h>

// Tensor geometry (compile-time constants from the reference).
#define NB   16           // batch
#define CH   64           // channels
#define HH   192          // height
#define WW   192          // width
#define PLANE (HH*WW)     // 36864 floats per (n,c) plane
#define QW   (WW/4)       // 48 float4 "col-quads" per row

typedef float v4f __attribute__((ext_vector_type(4)));

__device__ __forceinline__ v4f nt_load(const float* p) {
    return __builtin_nontemporal_load((const v4f*)p);
}
__device__ __forceinline__ void nt_store(float* p, v4f v) {
    __builtin_nontemporal_store(v, (v4f*)p);
}

// ---------------------------------------------------------------------------
// Segment 0: channels 0..15 (1x1 block => masked ReLU) and 56..63 (copy).
// One float4 per thread. 384 planes * 9216 quads = 3,538,944 units.
// ---------------------------------------------------------------------------
__global__ __launch_bounds__(256) void k_ew(const float* __restrict__ x,
                                            float* __restrict__ y) {
    const unsigned total = 384u * (PLANE / 4);
    unsigned u = blockIdx.x * blockDim.x + threadIdx.x;
    if (u >= total) return;
    unsigned p   = u / (PLANE / 4);
    unsigned off = u % (PLANE / 4);
    unsigned n   = p / 24;
    unsigned ci  = p % 24;
    unsigned c   = (ci < 16u) ? ci : (ci + 40u);   // 0..15 or 56..63
    size_t base  = ((size_t)n * CH + c) * PLANE + (size_t)off * 4;

    v4f v = nt_load(x + base);
    if (ci < 16u) {
        #pragma unroll
        for (int i = 0; i < 4; ++i) {
            float m = (v[i] > 0.0f) ? 1.0f : 0.0f;
            v[i] *= m;                              // matches xr*mask (keeps -0.0)
        }
    }
    nt_store(y + base, v);
}

// ---------------------------------------------------------------------------
// Segment 1: channels 16..31, 2x2 blocks. A thread owns a 2-row x 4-col tile
// = two complete 2x2 blocks; no cross-thread communication needed.
// 256 planes * 96 row-pairs * 48 quads = 1,179,648 units.
// ---------------------------------------------------------------------------
__global__ __launch_bounds__(256) void k_b2(const float* __restrict__ x,
                                            float* __restrict__ y) {
    const unsigned total = 256u * (HH / 2) * QW;
    unsigned u = blockIdx.x * blockDim.x + threadIdx.x;
    if (u >= total) return;
    unsigned q  = u % QW;
    unsigned rp = (u / QW) % (HH / 2);
    unsigned p  = u / (QW * (HH / 2));
    unsigned n  = p / 16;
    unsigned c  = 16u + (p % 16);
    size_t base = ((size_t)n * CH + c) * PLANE + (size_t)(rp * 2) * WW + q * 4;

    v4f a = nt_load(x + base);
    v4f b = nt_load(x + base + WW);
    float s01 = (a[0] + a[1]) + (b[0] + b[1]);
    float s23 = (a[2] + a[3]) + (b[2] + b[3]);
    float m01 = (s01 > 0.0f) ? 1.0f : 0.0f;
    float m23 = (s23 > 0.0f) ? 1.0f : 0.0f;
    a[0] *= m01; a[1] *= m01; b[0] *= m01; b[1] *= m01;
    a[2] *= m23; a[3] *= m23; b[2] *= m23; b[3] *= m23;
    nt_store(y + base, a);
    nt_store(y + base + WW, b);
}

// ---------------------------------------------------------------------------
// Segment 2: channels 32..47, 4x4 blocks. A thread owns exactly one block
// (4 rows x one float4). 256 planes * 48 * 48 = 589,824 units.
// ---------------------------------------------------------------------------
__global__ __launch_bounds__(256) void k_b4(const float* __restrict__ x,
                                            float* __restrict__ y) {
    const unsigned total = 256u * (HH / 4) * QW;
    unsigned u = blockIdx.x * blockDim.x + threadIdx.x;
    if (u >= total) return;
    unsigned q  = u % QW;
    unsigned rg = (u / QW) % (HH / 4);
    unsigned p  = u / (QW * (HH / 4));
    unsigned n  = p / 16;
    unsigned c  = 32u + (p % 16);
    size_t base = ((size_t)n * CH + c) * PLANE + (size_t)(rg * 4) * WW + q * 4;

    v4f r[4];
    float s = 0.0f;
    #pragma unroll
    for (int i = 0; i < 4; ++i) {
        r[i] = nt_load(x + base + (size_t)i * WW);
        s += (r[i][0] + r[i][1]) + (r[i][2] + r[i][3]);
    }
    float m = (s > 0.0f) ? 1.0f : 0.0f;
    #pragma unroll
    for (int i = 0; i < 4; ++i) {
        v4f v = r[i];
        v[0] *= m; v[1] *= m; v[2] *= m; v[3] *= m;
        nt_store(y + base + (size_t)i * WW, v);
    }
}

// ---------------------------------------------------------------------------
// Segment 3: channels 48..55, 8x8 blocks. A thread owns an 8-row x 4-col
// half-block; partner lane (lane^1) owns the other half. One wave32
// __shfl_xor combines the halves. Pairing is valid: global id parity equals
// lane parity (blockDim=256), and the unit count is even.
// 128 planes * 24 strips * 48 quads = 147,456 units.
// ---------------------------------------------------------------------------
__global__ __launch_bounds__(256) void k_b8(const float* __restrict__ x,
                                            float* __restrict__ y) {
    const unsigned total = 128u * (HH / 8) * QW;
    unsigned u = blockIdx.x * blockDim.x + threadIdx.x;
    if (u >= total) return;
    unsigned q  = u % QW;
    unsigned sg = (u / QW) % (HH / 8);
    unsigned p  = u / (QW * (HH / 8));
    unsigned n  = p / 8;
    unsigned c  = 48u + (p % 8);
    size_t base = ((size_t)n * CH + c) * PLANE + (size_t)(sg * 8) * WW + q * 4;

    v4f r[8];
    float s = 0.0f;
    #pragma unroll
    for (int i = 0; i < 8; ++i) {
        r[i] = nt_load(x + base + (size_t)i * WW);
        s += (r[i][0] + r[i][1]) + (r[i][2] + r[i][3]);
    }
    // Combine with partner half-block (cols 0..3 <-> cols 4..7 of same 8x8 block).
    s += __shfl_xor(s, 1, 32);
    float m = (s > 0.0f) ? 1.0f : 0.0f;
    #pragma unroll
    for (int i = 0; i < 8; ++i) {
        v4f v = r[i];
        v[0] *= m; v[1] *= m; v[2] *= m; v[3] *= m;
        nt_store(y + base + (size_t)i * WW, v);
    }
}

extern "C" void kernel_launch(void* const* d_in, const int* in_sizes, int n_in,
                              void* d_out, int out_size, void* d_ws, size_t ws_size,
                              hipStream_t stream) {
    (void)in_sizes; (void)n_in; (void)out_size; (void)d_ws; (void)ws_size;
    const float* x = (const float*)d_in[0];
    float* y = (float*)d_out;

    const unsigned t_ew = 384u * (PLANE / 4);          // 3,538,944
    const unsigned t_b2 = 256u * (HH / 2) * QW;        // 1,179,648
    const unsigned t_b4 = 256u * (HH / 4) * QW;        //   589,824
    const unsigned t_b8 = 128u * (HH / 8) * QW;        //   147,456

    k_ew<<<(t_ew + 255) / 256, 256, 0, stream>>>(x, y);
    k_b2<<<(t_b2 + 255) / 256, 256, 0, stream>>>(x, y);
    k_b4<<<(t_b4 + 255) / 256, 256, 0, stream>>>(x, y);
    k_b8<<<(t_b8 + 255) / 256, 256, 0, stream>>>(x, y);
}